// RAMAC_27771258536050
// MI455X (gfx1250) — compile-verified
//
#include <hip/hip_runtime.h>

typedef __attribute__((ext_vector_type(2))) float v2f;
typedef __attribute__((ext_vector_type(8))) float v8f;
typedef __attribute__((ext_vector_type(4))) int   v4i;

#define B_    16
#define C_    1024
#define HW_   4096
#define NPIX  65536      // B_*HW_
#define NREG  14
#define NCG   64         // channel groups of 16 channels
#define TROW  68         // LDS tile row stride (floats): 16B aligned + bank skew

#if __has_builtin(__builtin_amdgcn_global_load_async_to_lds_b128)
#define ASYNC_LDS 1
#endif

// col/row interval table: 0=[0,64) 1=[0,42) 2=[22,64) 3=[0,32) 4=[16,48) 5=[32,64)
__constant__ int   c_IS[6]      = {0, 0, 22, 0, 16, 32};
__constant__ int   c_IE[6]      = {64, 42, 64, 32, 48, 64};
__constant__ int   c_RI[NREG]   = {0, 1,1,2,2, 3,3,3,4,4,4,5,5,5};
__constant__ int   c_CI[NREG]   = {0, 1,2,1,2, 3,4,5,3,4,5,3,4,5};
__constant__ float c_AREA[NREG] = {4096.f, 1764.f,1764.f,1764.f,1764.f,
                                   1024.f,1024.f,1024.f,1024.f,1024.f,
                                   1024.f,1024.f,1024.f,1024.f};
// column segments {0,16,22,32,42,48,64}; interval q = segments [c_QS[q], c_QE[q])
__constant__ int   c_SEGB[7]    = {0, 16, 22, 32, 42, 48, 64};
__constant__ int   c_QS[6]      = {0, 0, 2, 0, 1, 3};
__constant__ int   c_QE[6]      = {6, 4, 6, 3, 5, 6};

__device__ __forceinline__ void wait_async_le(int n) {
#ifdef ASYNC_LDS
#if __has_builtin(__builtin_amdgcn_s_wait_asynccnt)
    if (n == 0) __builtin_amdgcn_s_wait_asynccnt(0);
    else        __builtin_amdgcn_s_wait_asynccnt(1);
#else
    if (n == 0) asm volatile("s_wait_asynccnt 0x0" ::: "memory");
    else        asm volatile("s_wait_asynccnt 0x1" ::: "memory");
#endif
#else
    (void)n;
#endif
}

#ifdef ASYNC_LDS
__device__ __forceinline__ void async_b128(const float* g, float* l) {
    __builtin_amdgcn_global_load_async_to_lds_b128(
        (__attribute__((address_space(1))) v4i*)(g),
        (__attribute__((address_space(3))) v4i*)(l), 0, 0);
}
#endif

__global__ void k_init(float* __restrict__ tot, int* __restrict__ cnt) {
    if (threadIdx.x == 0) tot[0] = 0.f;
    if (threadIdx.x < NREG) cnt[threadIdx.x] = 0;
}

// Fused streaming pass: block = (b, 16-channel group). Per row:
//  phase1: waves 0-3 -> WMMA 16x16x4 partial channel sums (A from LDS, B=ones)
//          threads 128..223 -> per-(channel,segment) maxes into LDS
//  phase2: threads 0..223 -> per-(channel,region) running maxes in registers
__global__ void __launch_bounds__(256)
k_fused(const float* __restrict__ x, float* __restrict__ sp, float* __restrict__ vt) {
    __shared__ float tile[2][16 * TROW];
    __shared__ float sgm[16 * 6];

    const int t    = threadIdx.x;
    const int b    = blockIdx.x >> 6;
    const int cg   = blockIdx.x & (NCG - 1);
    const int c0   = cg << 4;
    const int lch  = t >> 4;                 // loader: channel 0..15
    const int lq   = t & 15;                 // loader: float4 index in row
    const float* gbase = x + ((size_t)(b * C_ + c0 + lch)) * HW_ + lq * 4;
    float* spb = sp + ((size_t)(b * NCG + cg)) * HW_;

    const int lane = t & 31;
    const int wv   = t >> 5;
    const int lm   = lane & 15;
    const int kk   = (lane >> 4) << 1;

    const int p2ch = t & 15;                 // phase2: channel
    const int p2r  = t >> 4;                 // phase2: region (valid t<224)
    float regacc = -3.4e38f;

    v2f ones; ones.x = 1.f; ones.y = 1.f;

#ifdef ASYNC_LDS
    async_b128(gbase + 0 * 64, &tile[0][lch * TROW + lq * 4]);
    async_b128(gbase + 1 * 64, &tile[1][lch * TROW + lq * 4]);
#endif
    for (int h = 0; h < 64; ++h) {
        const int buf = h & 1;
#ifdef ASYNC_LDS
        wait_async_le(h < 63 ? 1 : 0);       // oldest (row h) complete
        __syncthreads();                     // all waves' row-h data in LDS
#else
        float4 v = *(const float4*)(gbase + h * 64);
        __syncthreads();                     // prev-row sgm readers done
        *(float4*)&tile[buf][lch * TROW + lq * 4] = v;
        __syncthreads();
#endif
        // ---------------- phase 1 ----------------
        if (t < 128) {                       // waves 0..3: WMMA channel sums
            const int pix0 = wv << 4;
            const float* tp = &tile[buf][0];
            v8f acc = {};
            for (int cc = 0; cc < 16; cc += 4) {
                v2f a;
                a.x = tp[(cc + kk) * TROW + pix0 + lm];
                a.y = tp[(cc + kk + 1) * TROW + pix0 + lm];
                acc = __builtin_amdgcn_wmma_f32_16x16x4_f32(
                    false, a, false, ones, (short)0, acc, false, false);
            }
            float* o = spb + h * 64 + pix0;  // D col 0: lane0 -> M=0..7, lane16 -> M=8..15
            if (lane == 0)  { for (int r = 0; r < 8; ++r) o[r]     = acc[r]; }
            if (lane == 16) { for (int r = 0; r < 8; ++r) o[8 + r] = acc[r]; }
        } else if (t < 224) {                // 96 threads: (channel, segment) maxes
            const int u  = t - 128;
            const int ch = u & 15;
            const int sg = u >> 4;
            const float* row = &tile[buf][ch * TROW];
            float mx = -3.4e38f;
            for (int w = c_SEGB[sg]; w < c_SEGB[sg + 1]; ++w) mx = fmaxf(mx, row[w]);
            sgm[ch * 6 + sg] = mx;
        }
        __syncthreads();                     // tile readers done; sgm ready
#ifdef ASYNC_LDS
        if (h + 2 < 64)                      // refill the freed buffer
            async_b128(gbase + (h + 2) * 64, &tile[buf][lch * TROW + lq * 4]);
#endif
        // ---------------- phase 2 ----------------
        if (t < 224) {
            const int ri = c_RI[p2r];
            if (h >= c_IS[ri] && h < c_IE[ri]) {
                const int q = c_CI[p2r];
                float cm = -3.4e38f;
                for (int sg = c_QS[q]; sg < c_QE[q]; ++sg)
                    cm = fmaxf(cm, sgm[p2ch * 6 + sg]);
                regacc = fmaxf(regacc, cm);
            }
        }
    }
    if (t < 224)
        vt[(size_t)p2r * (B_ * C_) + b * C_ + c0 + p2ch] = regacc;
}

// s[b,pix] = sum_cg s_part ; plus global total for the mean
__global__ void k_sum2(const float* __restrict__ sp, float* __restrict__ s,
                       float* __restrict__ tot) {
    __shared__ float sh[256];
    int t = threadIdx.x;
    int p = blockIdx.x * 256 + t;
    int b = p >> 12, pix = p & (HW_ - 1);
    const float* q = sp + (size_t)b * NCG * HW_ + pix;
    float v = 0.f;
    for (int cg = 0; cg < NCG; ++cg) v += q[(size_t)cg * HW_];
    s[p] = v;
    sh[t] = v;
    __syncthreads();
    for (int o = 128; o > 0; o >>= 1) {
        if (t < o) sh[t] += sh[t + o];
        __syncthreads();
    }
    if (t == 0) atomicAdd(tot, sh[0]);
}

// per-region counts of tt = (s - mean > 0) over all batches
__global__ void k_count(const float* __restrict__ s, const float* __restrict__ tot,
                        int* __restrict__ cnt) {
    __shared__ int sc[NREG];
    int t = threadIdx.x;
    if (t < NREG) sc[t] = 0;
    __syncthreads();
    int p    = blockIdx.x * 256 + t;
    float mn = tot[0] * (1.f / NPIX);
    int pix  = p & (HW_ - 1);
    int h    = pix >> 6, w = pix & 63;
    if (s[p] - mn > 0.f) {
        for (int r = 0; r < NREG; ++r) {
            int ri = c_RI[r], ci = c_CI[r];
            if (h >= c_IS[ri] && h < c_IE[ri] && w >= c_IS[ci] && w < c_IE[ci])
                atomicAdd(&sc[r], 1);
        }
    }
    __syncthreads();
    if (t < NREG) atomicAdd(&cnt[t], sc[t]);
}

// nrm[b,r] = || vt[r][b][:] ||_2
__global__ void k_nrm(const float* __restrict__ vt, float* __restrict__ nrm) {
    __shared__ float sh[256];
    int t = threadIdx.x;
    int b = blockIdx.x / NREG;
    int r = blockIdx.x % NREG;
    const float* vp = vt + (size_t)r * (B_ * C_) + b * C_;
    float v0 = vp[t], v1 = vp[t + 256], v2 = vp[t + 512], v3 = vp[t + 768];
    sh[t] = v0 * v0 + v1 * v1 + v2 * v2 + v3 * v3;
    __syncthreads();
    for (int o = 128; o > 0; o >>= 1) {
        if (t < o) sh[t] += sh[t + o];
        __syncthreads();
    }
    if (t == 0) nrm[blockIdx.x] = sqrtf(sh[0]);
}

// out[b,c] = sum_r mult_r * wgt_r * vt[r][b][c] / (nrm[b,r] + eps)
__global__ void k_out(const float* __restrict__ vt, const float* __restrict__ nrm,
                      const int* __restrict__ cnt, float* __restrict__ out) {
    int i = blockIdx.x * 256 + threadIdx.x;
    int b = i >> 10;
    float acc = 0.f;
    for (int r = 0; r < NREG; ++r) {
        float wgt = (float)cnt[r] / c_AREA[r];
        if (wgt <= (1.f / 3.f)) wgt = 0.f;
        float mult = (r == 0) ? 2.f : 1.f;   // full region appears twice in reference
        acc += mult * wgt * vt[(size_t)r * (B_ * C_) + i] / (nrm[b * NREG + r] + 1e-6f);
    }
    out[i] = acc;
}

extern "C" void kernel_launch(void* const* d_in, const int* in_sizes, int n_in,
                              void* d_out, int out_size, void* d_ws, size_t ws_size,
                              hipStream_t stream) {
    (void)in_sizes; (void)n_in; (void)out_size; (void)ws_size;
    const float* x = (const float*)d_in[0];
    float* out = (float*)d_out;

    float* wsf  = (float*)d_ws;
    float* tot  = wsf;                               // 1 float
    int*   cnt  = (int*)(wsf + 4);                   // 14 ints
    float* s    = wsf + 32;                          // 65536 floats
    float* vt   = wsf + 32 + NPIX;                   // 14*16384 floats
    float* nrm  = vt + (size_t)NREG * B_ * C_;       // 224 floats
    float* spart= nrm + 256;                         // 16*64*4096 floats (16 MB)

    k_init <<<1, 32, 0, stream>>>(tot, cnt);
    k_fused<<<B_ * NCG, 256, 0, stream>>>(x, spart, vt);     // 1024 blocks
    k_sum2 <<<NPIX / 256, 256, 0, stream>>>(spart, s, tot);
    k_count<<<NPIX / 256, 256, 0, stream>>>(s, tot, cnt);
    k_nrm  <<<B_ * NREG, 256, 0, stream>>>(vt, nrm);
    k_out  <<<(B_ * C_) / 256, 256, 0, stream>>>(vt, nrm, cnt, out);
}